// HGResidualBlock_730144440554
// MI455X (gfx1250) — compile-verified
//
#include <hip/hip_runtime.h>
#include <cstddef>

typedef float v2f __attribute__((ext_vector_type(2)));
typedef float v8f __attribute__((ext_vector_type(8)));
typedef unsigned int u32x4 __attribute__((ext_vector_type(4)));
typedef int i32x4 __attribute__((ext_vector_type(4)));
typedef int i32x8 __attribute__((ext_vector_type(8)));

#define EPS 1e-5f

#if __has_builtin(__builtin_amdgcn_tensor_load_to_lds) && \
    __has_builtin(__builtin_amdgcn_s_wait_tensorcnt)
#define USE_TDM 1
#else
#define USE_TDM 0
#endif

// ---------------------------------------------------------------------------
// TDM helpers: issue a 2-D tile DMA (global -> LDS) with pad-on-store.
// width elems per row (multiple of 32), LDS row stride = width + 4*(width/32)
// (pad_interval = 32 dwords, pad_amount = 4 dwords  ->  stride 36 / 144).
// Descriptor packing per CDNA5 ISA 8.3/8.4 (D# group0 / group1).
// ---------------------------------------------------------------------------
#if USE_TDM
__device__ __forceinline__ unsigned lds_byte_off(const void* p) {
  // flat LDS addresses carry the group-segment byte offset in addr[31:0]
  return (unsigned)(unsigned long long)(uintptr_t)p;
}

__device__ __forceinline__ void tdm_load_tile_2d(unsigned lds_off,
                                                 const float* g,
                                                 int width, int rows,
                                                 int row_stride_elems) {
  unsigned long long ga = (unsigned long long)(uintptr_t)g;
  u32x4 g0;
  g0[0] = 1u;                                    // count=1, user mode
  g0[1] = lds_off;                               // lds_addr (bytes)
  g0[2] = (unsigned)ga;                          // global_addr[31:0]
  g0[3] = (unsigned)(ga >> 32) | (2u << 30);     // global_addr[56:32], type=2
  i32x8 g1;
  g1[0] = (2 << 16)      // data_size = 4 bytes
        | (1 << 20)      // pad_enable
        | (4 << 22)      // pad_interval: 32 dwords
        | (3 << 25);     // pad_amount: 4 dwords
  g1[1] = (width & 0xffff) << 16;                // tensor_dim0[15:0]
  g1[2] = ((width >> 16) & 0xffff) | ((rows & 0xffff) << 16);  // dim0 hi | dim1 lo
  g1[3] = ((rows >> 16) & 0xffff) | ((width & 0xffff) << 16);  // dim1 hi | tile_dim0
  g1[4] = rows & 0xffff;                         // tile_dim1 | tile_dim2=0
  g1[5] = row_stride_elems;                      // tensor_dim0_stride[31:0]
  g1[6] = 0;
  g1[7] = 0;
  i32x4 z4 = {0, 0, 0, 0};
#if defined(__clang_major__) && (__clang_major__ >= 23)
  i32x8 z8 = {0, 0, 0, 0, 0, 0, 0, 0};
  __builtin_amdgcn_tensor_load_to_lds(g0, g1, z4, z4, z8, 0);
#else
  __builtin_amdgcn_tensor_load_to_lds(g0, g1, z4, z4, 0);
#endif
}
#endif  // USE_TDM

// ---------------------------------------------------------------------------
// Shared WMMA inner chunk: 32 K-steps from padded (stride 36) LDS tiles.
// A-frag: lanes 0-15 K-pair {0,1}, lanes 16-31 {2,3} (ISA 7.12.2).
// ---------------------------------------------------------------------------
__device__ __forceinline__ void mma_chunk36(const float* __restrict__ cA,
                                            const float* __restrict__ cW,
                                            int wm, int wn, int rl, int koff,
                                            v8f acc[2][2]) {
#pragma unroll
  for (int kk = 0; kk < 32; kk += 4) {
    v2f a[2], b[2];
#pragma unroll
    for (int mt = 0; mt < 2; ++mt) {
      const float* p = &cA[(wm * 32 + mt * 16 + rl) * 36 + kk + koff];
      a[mt] = (v2f){p[0], p[1]};
    }
#pragma unroll
    for (int nt = 0; nt < 2; ++nt) {
      const float* p = &cW[(wn * 32 + nt * 16 + rl) * 36 + kk + koff];
      b[nt] = (v2f){p[0], p[1]};
    }
#pragma unroll
    for (int mt = 0; mt < 2; ++mt)
#pragma unroll
      for (int nt = 0; nt < 2; ++nt)
        acc[mt][nt] = __builtin_amdgcn_wmma_f32_16x16x4_f32(
            false, a[mt], false, b[nt], (short)0, acc[mt][nt], false, false);
  }
}

// ---------------------------------------------------------------------------
// K0: fold the 9 horizontal replicas of Wg:  Wr[j,d] = sum_r Wg[j, r*256+d]
// ---------------------------------------------------------------------------
__global__ __launch_bounds__(256) void wg_reduce(const float* __restrict__ Wg,
                                                 float* __restrict__ Wr) {
  int idx = blockIdx.x * 256 + threadIdx.x;
  int j = idx >> 8, d = idx & 255;
  const float* p = Wg + (size_t)j * 2304 + d;
  float s = 0.f;
#pragma unroll
  for (int r = 0; r < 9; ++r) s += p[r * 256];
  Wr[idx] = s;
}

// ---------------------------------------------------------------------------
// NT GEMM via V_WMMA_F32_16X16X4_F32:  Y[MxN] = X[MxK] * W[NxK]^T
// 256 threads / 8 waves, 64(M) x 128(N) block tile, K-chunks of 32.
// TDM path: double-buffered LDS, wave 0 drives the tensor DMA.
// ---------------------------------------------------------------------------
__global__ __launch_bounds__(256) void gemm_nt(const float* __restrict__ X,
                                               const float* __restrict__ W,
                                               float* __restrict__ Y,
                                               int N, int K) {
  const int tid  = threadIdx.x;
  const int lane = tid & 31;
  const int wave = tid >> 5;
  const int wm   = wave & 1;
  const int wn   = wave >> 1;
  const int m0   = blockIdx.x * 64;
  const int n0   = blockIdx.y * 128;
  const int rl   = lane & 15;
  const int koff = (lane >> 4) << 1;
  const int rhi  = (lane >> 4) << 3;

  v8f acc[2][2] = {};

#if USE_TDM
  __shared__ alignas(16) float sA[2 * 64 * 36];
  __shared__ alignas(16) float sW[2 * 128 * 36];
  const int nch = K >> 5;
  if (wave == 0) {
    tdm_load_tile_2d(lds_byte_off(sA), X + (size_t)m0 * K, 32, 64, K);
    tdm_load_tile_2d(lds_byte_off(sW), W + (size_t)n0 * K, 32, 128, K);
  }
  for (int c = 0; c < nch; ++c) {
    const float* cA = sA + (c & 1) * (64 * 36);
    const float* cW = sW + (c & 1) * (128 * 36);
    if (wave == 0) {
      if (c + 1 < nch) {
        int nb = (c + 1) & 1;
        tdm_load_tile_2d(lds_byte_off(sA) + nb * 64 * 36 * 4,
                         X + (size_t)m0 * K + (c + 1) * 32, 32, 64, K);
        tdm_load_tile_2d(lds_byte_off(sW) + nb * 128 * 36 * 4,
                         W + (size_t)n0 * K + (c + 1) * 32, 32, 128, K);
        __builtin_amdgcn_s_wait_tensorcnt(2);
      } else {
        __builtin_amdgcn_s_wait_tensorcnt(0);
      }
    }
    __syncthreads();
    mma_chunk36(cA, cW, wm, wn, rl, koff, acc);
    __syncthreads();
  }
#else
  __shared__ alignas(16) float sA[64 * 36];
  __shared__ alignas(16) float sW[128 * 36];
  for (int kt = 0; kt < K; kt += 32) {
#pragma unroll
    for (int i = 0; i < 2; ++i) {
      int q = tid + i * 256;
      int r = q >> 3, c4 = (q & 7) << 2;
      float4 v = *(const float4*)(X + (size_t)(m0 + r) * K + kt + c4);
      *(float4*)(&sA[r * 36 + c4]) = v;
    }
#pragma unroll
    for (int i = 0; i < 4; ++i) {
      int q = tid + i * 256;
      int r = q >> 3, c4 = (q & 7) << 2;
      float4 v = *(const float4*)(W + (size_t)(n0 + r) * K + kt + c4);
      *(float4*)(&sW[r * 36 + c4]) = v;
    }
    __syncthreads();
    mma_chunk36(sA, sW, wm, wn, rl, koff, acc);
    __syncthreads();
  }
#endif

#pragma unroll
  for (int mt = 0; mt < 2; ++mt)
#pragma unroll
    for (int nt = 0; nt < 2; ++nt)
#pragma unroll
      for (int v = 0; v < 8; ++v) {
        int row = m0 + wm * 32 + mt * 16 + v + rhi;
        int col = n0 + wn * 32 + nt * 16 + rl;
        Y[(size_t)row * N + col] = acc[mt][nt][v];
      }
}

// ---------------------------------------------------------------------------
// K5: per-batch coef contraction:
//   Y3[b] (64x256) = coef[b] viewed (64 x 576) @ Y2[b] viewed (576 x 256)
// Grid: (2 n-tiles, 256 batches). A staged stride-36; B stride 132 (sync)
// or 144 (TDM pads every 32 dwords of the 128-wide row).
// ---------------------------------------------------------------------------
__global__ __launch_bounds__(256) void coef_gemm(const float* __restrict__ coef,
                                                 const float* __restrict__ Y2,
                                                 float* __restrict__ Y3) {
  const int tid  = threadIdx.x;
  const int lane = tid & 31;
  const int wave = tid >> 5;
  const int wm   = wave & 1;
  const int wn   = wave >> 1;
  const int b    = blockIdx.y;
  const int n0   = blockIdx.x * 128;
  const int rl   = lane & 15;
  const int koff = (lane >> 4) << 1;
  const int rhi  = (lane >> 4) << 3;

  const float* cb = coef + (size_t)b * 9 * 64 * 64;  // [k][g][h]
  const float* Bb = Y2 + (size_t)b * 147456;         // [576][256] contiguous

  v8f acc[2][2] = {};

#if USE_TDM
  __shared__ alignas(16) float sA[2 * 64 * 36];
  __shared__ alignas(16) float sB[2 * 32 * 144];   // 128-wide rows, stride 144
  const int nch = 18;  // 576 / 32
  if (wave == 0) {
    tdm_load_tile_2d(lds_byte_off(sA), cb, 32, 64, 64);
    tdm_load_tile_2d(lds_byte_off(sB), Bb + n0, 128, 32, 256);
  }
  for (int c = 0; c < nch; ++c) {
    const float* cA = sA + (c & 1) * (64 * 36);
    const float* cB = sB + (c & 1) * (32 * 144);
    if (wave == 0) {
      if (c + 1 < nch) {
        int kt = (c + 1) * 32;
        int nb = (c + 1) & 1;
        // A: coef[b, kt/64, g, kt%64 .. +31]  (row stride 64)
        tdm_load_tile_2d(lds_byte_off(sA) + nb * 64 * 36 * 4,
                         cb + (kt >> 6) * 4096 + (kt & 63), 32, 64, 64);
        tdm_load_tile_2d(lds_byte_off(sB) + nb * 32 * 144 * 4,
                         Bb + (size_t)kt * 256 + n0, 128, 32, 256);
        __builtin_amdgcn_s_wait_tensorcnt(2);
      } else {
        __builtin_amdgcn_s_wait_tensorcnt(0);
      }
    }
    __syncthreads();
#pragma unroll
    for (int kk = 0; kk < 32; kk += 4) {
      v2f a[2], bf[2];
#pragma unroll
      for (int mt = 0; mt < 2; ++mt) {
        const float* p = &cA[(wm * 32 + mt * 16 + rl) * 36 + kk + koff];
        a[mt] = (v2f){p[0], p[1]};
      }
#pragma unroll
      for (int nt = 0; nt < 2; ++nt) {
        int col = wn * 32 + nt * 16;
        int colp = col + ((col >> 5) << 2) + rl;   // +4 pad per 32-dword group
        bf[nt] = (v2f){cB[(kk + koff) * 144 + colp],
                       cB[(kk + koff + 1) * 144 + colp]};
      }
#pragma unroll
      for (int mt = 0; mt < 2; ++mt)
#pragma unroll
        for (int nt = 0; nt < 2; ++nt)
          acc[mt][nt] = __builtin_amdgcn_wmma_f32_16x16x4_f32(
              false, a[mt], false, bf[nt], (short)0, acc[mt][nt], false, false);
    }
    __syncthreads();
  }
#else
  __shared__ alignas(16) float sA[64 * 36];
  __shared__ alignas(16) float sB[32 * 132];
  for (int kt = 0; kt < 576; kt += 32) {
    const int kblk = kt >> 6;
    const int h0   = kt & 63;
#pragma unroll
    for (int i = 0; i < 2; ++i) {
      int q = tid + i * 256;
      int g = q >> 3, c4 = (q & 7) << 2;
      float4 v = *(const float4*)(cb + kblk * 4096 + g * 64 + h0 + c4);
      *(float4*)(&sA[g * 36 + c4]) = v;
    }
#pragma unroll
    for (int i = 0; i < 4; ++i) {
      int q = tid + i * 256;
      int r = q >> 5, c4 = (q & 31) << 2;
      float4 v = *(const float4*)(Bb + (size_t)(kt + r) * 256 + n0 + c4);
      *(float4*)(&sB[r * 132 + c4]) = v;
    }
    __syncthreads();
#pragma unroll
    for (int kk = 0; kk < 32; kk += 4) {
      v2f a[2], bf[2];
#pragma unroll
      for (int mt = 0; mt < 2; ++mt) {
        const float* p = &sA[(wm * 32 + mt * 16 + rl) * 36 + kk + koff];
        a[mt] = (v2f){p[0], p[1]};
      }
#pragma unroll
      for (int nt = 0; nt < 2; ++nt) {
        int col = wn * 32 + nt * 16 + rl;
        bf[nt] = (v2f){sB[(kk + koff) * 132 + col],
                       sB[(kk + koff + 1) * 132 + col]};
      }
#pragma unroll
      for (int mt = 0; mt < 2; ++mt)
#pragma unroll
        for (int nt = 0; nt < 2; ++nt)
          acc[mt][nt] = __builtin_amdgcn_wmma_f32_16x16x4_f32(
              false, a[mt], false, bf[nt], (short)0, acc[mt][nt], false, false);
    }
    __syncthreads();
  }
#endif

  float* out = Y3 + (size_t)b * 64 * 256;
#pragma unroll
  for (int mt = 0; mt < 2; ++mt)
#pragma unroll
    for (int nt = 0; nt < 2; ++nt)
#pragma unroll
      for (int v = 0; v < 8; ++v) {
        int row = wm * 32 + mt * 16 + v + rhi;
        int col = n0 + wn * 32 + nt * 16 + rl;
        out[(size_t)row * 256 + col] = acc[mt][nt][v];
      }
}

// ---------------------------------------------------------------------------
// BN stats: one block per channel g, stats over 256*256 elems (deterministic)
// ---------------------------------------------------------------------------
__global__ __launch_bounds__(256) void bn_stats(const float* __restrict__ X,
                                                float* __restrict__ stats) {
  const int g = blockIdx.x;
  const int tid = threadIdx.x;
  float s = 0.f, sq = 0.f;
  for (int i = tid; i < 256 * 256; i += 256) {
    int bb = i >> 8, d = i & 255;
    float x = X[(size_t)bb * 16384 + g * 256 + d];
    s += x;
    sq += x * x;
  }
  __shared__ float rs[256], rq[256];
  rs[tid] = s;
  rq[tid] = sq;
  __syncthreads();
  for (int off = 128; off > 0; off >>= 1) {
    if (tid < off) {
      rs[tid] += rs[tid + off];
      rq[tid] += rq[tid + off];
    }
    __syncthreads();
  }
  if (tid == 0) {
    float mean = rs[0] * (1.0f / 65536.0f);
    float var = rq[0] * (1.0f / 65536.0f) - mean * mean;
    stats[g * 2] = mean;
    stats[g * 2 + 1] = rsqrtf(var + EPS);
  }
}

// ---------------------------------------------------------------------------
// BN apply (+ optional residual, + optional ReLU)
// ---------------------------------------------------------------------------
__global__ __launch_bounds__(256) void bn_apply(const float* __restrict__ X,
                                                const float* __restrict__ stats,
                                                const float* __restrict__ gamma,
                                                const float* __restrict__ beta,
                                                const float* __restrict__ residual,
                                                float* __restrict__ out,
                                                int do_relu) {
  size_t idx = (size_t)blockIdx.x * 256 + threadIdx.x;
  int g = ((int)(idx >> 8)) & 63;
  float mean = stats[g * 2], rstd = stats[g * 2 + 1];
  float y = (X[idx] - mean) * rstd * gamma[g] + beta[g];
  if (residual) y += residual[idx];
  if (do_relu) y = fmaxf(y, 0.f);
  out[idx] = y;
}

// ---------------------------------------------------------------------------
// Orchestration. d_out doubles as the Y4 scratch. Workspace ~187 MB f32.
// ---------------------------------------------------------------------------
extern "C" void kernel_launch(void* const* d_in, const int* in_sizes, int n_in,
                              void* d_out, int out_size, void* d_ws, size_t ws_size,
                              hipStream_t stream) {
  const float* inp  = (const float*)d_in[0];
  const float* coef = (const float*)d_in[1];
  const float* W1   = (const float*)d_in[2];
  const float* Wg   = (const float*)d_in[3];
  const float* W2   = (const float*)d_in[4];
  const float* g1   = (const float*)d_in[5];
  const float* b1   = (const float*)d_in[6];
  const float* g2   = (const float*)d_in[7];
  const float* b2   = (const float*)d_in[8];
  const float* g3   = (const float*)d_in[9];
  const float* b3   = (const float*)d_in[10];
  float* out = (float*)d_out;

  float* ws  = (float*)d_ws;
  float* Wr  = ws;                   // 589824
  float* X1  = Wr + 589824;          // 4194304
  float* Y2  = X1 + 4194304;         // 37748736
  float* Y3  = Y2 + 37748736;        // 4194304
  float* st1 = Y3 + 4194304;         // 128
  float* st2 = st1 + 128;            // 128
  float* st3 = st2 + 128;            // 128

  wg_reduce<<<2304, 256, 0, stream>>>(Wg, Wr);

  gemm_nt<<<dim3(256, 2), 256, 0, stream>>>(inp, W1, X1, 256, 256);
  bn_stats<<<64, 256, 0, stream>>>(X1, st1);
  bn_apply<<<16384, 256, 0, stream>>>(X1, st1, g1, b1, nullptr, X1, 1);

  gemm_nt<<<dim3(256, 18), 256, 0, stream>>>(X1, Wr, Y2, 2304, 256);
  coef_gemm<<<dim3(2, 256), 256, 0, stream>>>(coef, Y2, Y3);
  bn_stats<<<64, 256, 0, stream>>>(Y3, st2);
  bn_apply<<<16384, 256, 0, stream>>>(Y3, st2, g2, b2, nullptr, Y3, 1);

  gemm_nt<<<dim3(256, 2), 256, 0, stream>>>(Y3, W2, out, 256, 256);
  bn_stats<<<64, 256, 0, stream>>>(out, st3);
  bn_apply<<<16384, 256, 0, stream>>>(out, st3, g3, b3, inp, out, 1);
}